// Model_2310692406033
// MI455X (gfx1250) — compile-verified
//
#include <hip/hip_runtime.h>
#include <hip/hip_bf16.h>

typedef __attribute__((ext_vector_type(2))) float v2f;
typedef __attribute__((ext_vector_type(8))) float v8f;

#define N_SRC1 500000
#define N_DST1 100000
#define N_DST2 20000
#define IN_F   128
#define H_F    256
#define N_CLS  47

// ---------------------------------------------------------------------------
// Zero a contiguous float region (float4 grid-stride).
// ---------------------------------------------------------------------------
__global__ __launch_bounds__(256) void zero_kernel(float4* __restrict__ p, long n4) {
    long i = (long)blockIdx.x * blockDim.x + threadIdx.x;
    long stride = (long)gridDim.x * blockDim.x;
    float4 z = make_float4(0.f, 0.f, 0.f, 0.f);
    for (; i < n4; i += stride) p[i] = z;
}

// ---------------------------------------------------------------------------
// Scatter-add layer 1: one wave per edge, feature dim 128 (4 floats/lane).
// ---------------------------------------------------------------------------
__global__ __launch_bounds__(256) void scatter1_kernel(const float* __restrict__ x,
                                                       const int* __restrict__ src,
                                                       const int* __restrict__ dst,
                                                       float* __restrict__ agg,
                                                       float* __restrict__ deg,
                                                       int nedges) {
    int e = (int)(((long)blockIdx.x * blockDim.x + threadIdx.x) >> 5);
    if (e >= nedges) return;
    int lane = threadIdx.x & 31;
    int s = src[e], d = dst[e];
    const float4* xr = (const float4*)(x + (size_t)s * IN_F);
    float4 v = xr[lane];                    // 16B coalesced gather
    float* ar = agg + (size_t)d * IN_F + lane * 4;
    atomicAdd(ar + 0, v.x);
    atomicAdd(ar + 1, v.y);
    atomicAdd(ar + 2, v.z);
    atomicAdd(ar + 3, v.w);
    if (lane == 0) atomicAdd(deg + d, 1.0f);
}

// ---------------------------------------------------------------------------
// Scatter-add layer 2: one wave per edge, feature dim 256 (8 floats/lane).
// ---------------------------------------------------------------------------
__global__ __launch_bounds__(256) void scatter2_kernel(const float* __restrict__ h,
                                                       const int* __restrict__ src,
                                                       const int* __restrict__ dst,
                                                       float* __restrict__ agg,
                                                       float* __restrict__ deg,
                                                       int nedges) {
    int e = (int)(((long)blockIdx.x * blockDim.x + threadIdx.x) >> 5);
    if (e >= nedges) return;
    int lane = threadIdx.x & 31;
    int s = src[e], d = dst[e];
    const float4* hr = (const float4*)(h + (size_t)s * H_F);
    float4 v0 = hr[lane * 2 + 0];
    float4 v1 = hr[lane * 2 + 1];
    float* ar = agg + (size_t)d * H_F + lane * 8;
    atomicAdd(ar + 0, v0.x); atomicAdd(ar + 1, v0.y);
    atomicAdd(ar + 2, v0.z); atomicAdd(ar + 3, v0.w);
    atomicAdd(ar + 4, v1.x); atomicAdd(ar + 5, v1.y);
    atomicAdd(ar + 6, v1.z); atomicAdd(ar + 7, v1.w);
    if (lane == 0) atomicAdd(deg + d, 1.0f);
}

// ---------------------------------------------------------------------------
// Layer-1 fused GEMM:  h[i] = relu( [x_i | agg_i/max(deg,1)] @ [Ws; Wn] + b )
//   M = 100000, K = 256 (chunk 0 = x @ Wself, chunk 1 = agg @ Wneigh), N = 256.
// Block = 8 waves; each wave owns one 16x64 output strip (4 x v8f accum).
// LDS holds the 128x64 weight chunk in FRAGMENT-PACKED order:
//   element (kk, nn) -> Bs[(kk>>1)*PK1 + nn*2 + (kk&1)]
// so each B-fragment {B[kb][col], B[kb+1][col]} is one ds_load_b64.
// Row stride PK1=160 (== 32 mod 64 banks) keeps the two lane-halves
// (which read adjacent k-pair rows) on disjoint bank halves.
// ---------------------------------------------------------------------------
#define MT1 (N_DST1 / 16)    /* 6250 */
#define NB1 (H_F / 64)       /* 4 N-blocks of 64 */
#define PK1 160

__global__ __launch_bounds__(256) void gemm1_kernel(const float* __restrict__ x,
                                                    const float* __restrict__ agg,
                                                    const float* __restrict__ deg,
                                                    const float* __restrict__ Wself,
                                                    const float* __restrict__ Wneigh,
                                                    const float* __restrict__ bias,
                                                    float* __restrict__ h) {
    __shared__ float Bs[64 * PK1];           // 40 KB packed K-chunk x N-strip
    const int warp = threadIdx.x >> 5;
    const int lane = threadIdx.x & 31;
    const int half = lane >> 4;              // lanes 16-31 hold K+2
    const int l15  = lane & 15;
    const int nb = blockIdx.x % NB1;
    const int mt = (blockIdx.x / NB1) * 8 + warp;
    const int ntbase = nb * 64;
    const bool active = (mt < MT1);          // wave-uniform

    const int row = mt * 16 + l15;
    const float* xrow = x + (size_t)row * IN_F;
    const float* arow = agg + (size_t)row * IN_F;
    float invd = 1.0f;
    if (active) invd = 1.0f / fmaxf(deg[row], 1.0f);

    v8f acc[4] = {};
    for (int kc = 0; kc < 2; ++kc) {
        // ---- stage 128x64 chunk, interleaving K-pairs (b128 in, b128 out) ----
        const float4* Wv = (const float4*)((kc == 0) ? Wself : Wneigh);
        for (int it = threadIdx.x; it < 64 * 16; it += 256) {
            int kp = it >> 4, nn4 = it & 15;
            float4 r0 = Wv[(2 * kp)     * (H_F / 4) + (ntbase >> 2) + nn4];
            float4 r1 = Wv[(2 * kp + 1) * (H_F / 4) + (ntbase >> 2) + nn4];
            float4* dp = (float4*)&Bs[kp * PK1 + nn4 * 8];
            dp[0] = make_float4(r0.x, r1.x, r0.y, r1.y);
            dp[1] = make_float4(r0.z, r1.z, r0.w, r1.w);
        }
        __syncthreads();

        if (active) {
            const float* A = (kc == 0) ? xrow : arow;
            const float s = (kc == 0) ? 1.0f : invd;
#pragma unroll 4
            for (int k = 0; k < 128; k += 4) {
                v2f a = *(const v2f*)(A + k + 2 * half);
                a *= s;                              // v_pk_mul_f32
                const float* brow = &Bs[((k >> 1) + half) * PK1 + l15 * 2];
#pragma unroll
                for (int t = 0; t < 4; ++t) {
                    v2f b = *(const v2f*)(brow + t * 32);   // ds_load_b64
                    acc[t] = __builtin_amdgcn_wmma_f32_16x16x4_f32(
                        false, a, false, b, (short)0, acc[t], false, false);
                }
            }
        }
        __syncthreads();
    }

    if (active) {
#pragma unroll
        for (int t = 0; t < 4; ++t) {
            const int col = ntbase + t * 16 + l15;
            const float bv = bias[col];
#pragma unroll
            for (int v = 0; v < 8; ++v) {
                int r = mt * 16 + v + 8 * half;
                h[(size_t)r * H_F + col] = fmaxf(acc[t][v] + bv, 0.0f);
            }
        }
    }
}

// ---------------------------------------------------------------------------
// Layer-2 fused GEMM:  out[i] = [h_i | agg2_i/max(deg,1)] @ [Ws2; Wn2] + b2
//   M = 20000, K = 512 in 4 chunks of 128, N = 47 (3 tiles, padded to 48).
// Packed row stride 96 floats (== 32 mod 64 banks): conflict-free as-is.
// ---------------------------------------------------------------------------
#define MT2 (N_DST2 / 16)    /* 1250 */
#define PK2 96

__global__ __launch_bounds__(256) void gemm2_kernel(const float* __restrict__ h,
                                                    const float* __restrict__ agg,
                                                    const float* __restrict__ deg,
                                                    const float* __restrict__ Wself,
                                                    const float* __restrict__ Wneigh,
                                                    const float* __restrict__ bias,
                                                    float* __restrict__ out) {
    __shared__ float Bs[64 * PK2];           // 24 KB packed K-chunk x N(48)
    const int warp = threadIdx.x >> 5;
    const int lane = threadIdx.x & 31;
    const int half = lane >> 4;
    const int l15  = lane & 15;
    const int mt = blockIdx.x * 8 + warp;
    const bool active = (mt < MT2);

    const int row = mt * 16 + l15;
    const float* hrow = h + (size_t)row * H_F;
    const float* arow = agg + (size_t)row * H_F;
    float invd = 1.0f;
    if (active) invd = 1.0f / fmaxf(deg[row], 1.0f);

    v8f acc[3] = {};
    for (int kc = 0; kc < 4; ++kc) {
        const float* W = (kc < 2) ? Wself : Wneigh;
        const int krow0 = (kc & 1) * 128;
        for (int it = threadIdx.x; it < 64 * 48; it += 256) {
            int kp = it / 48, nn = it % 48;
            v2f pr; pr.x = 0.0f; pr.y = 0.0f;
            if (nn < N_CLS) {
                pr.x = W[(size_t)(krow0 + 2 * kp)     * N_CLS + nn];
                pr.y = W[(size_t)(krow0 + 2 * kp + 1) * N_CLS + nn];
            }
            *(v2f*)&Bs[kp * PK2 + nn * 2] = pr;
        }
        __syncthreads();

        if (active) {
            const float* A = ((kc < 2) ? hrow : arow) + krow0;
            const float s = (kc < 2) ? 1.0f : invd;
#pragma unroll 4
            for (int k = 0; k < 128; k += 4) {
                v2f a = *(const v2f*)(A + k + 2 * half);
                a *= s;
                const float* brow = &Bs[((k >> 1) + half) * PK2 + l15 * 2];
#pragma unroll
                for (int t = 0; t < 3; ++t) {
                    v2f b = *(const v2f*)(brow + t * 32);   // ds_load_b64
                    acc[t] = __builtin_amdgcn_wmma_f32_16x16x4_f32(
                        false, a, false, b, (short)0, acc[t], false, false);
                }
            }
        }
        __syncthreads();
    }

    if (active) {
#pragma unroll
        for (int t = 0; t < 3; ++t) {
            const int col = t * 16 + l15;
            if (col < N_CLS) {               // divergent only in the epilogue
                const float bv = bias[col];
#pragma unroll
                for (int v = 0; v < 8; ++v) {
                    int r = mt * 16 + v + 8 * half;
                    out[(size_t)r * N_CLS + col] = acc[t][v] + bv;
                }
            }
        }
    }
}

// ---------------------------------------------------------------------------
// Launch
// ---------------------------------------------------------------------------
extern "C" void kernel_launch(void* const* d_in, const int* in_sizes, int n_in,
                              void* d_out, int out_size, void* d_ws, size_t ws_size,
                              hipStream_t stream) {
    const float* x      = (const float*)d_in[0];
    const float* Wself1 = (const float*)d_in[1];
    const float* Wneigh1= (const float*)d_in[2];
    const float* b1     = (const float*)d_in[3];
    const float* Wself2 = (const float*)d_in[4];
    const float* Wneigh2= (const float*)d_in[5];
    const float* b2     = (const float*)d_in[6];
    const int*   src1   = (const int*)d_in[7];
    const int*   dst1   = (const int*)d_in[8];
    const int*   src2   = (const int*)d_in[9];
    const int*   dst2   = (const int*)d_in[10];
    const int E1 = in_sizes[7];
    const int E2 = in_sizes[9];
    float* out = (float*)d_out;

    // Workspace layout (floats): agg1 | deg1 | agg2 | deg2 | h
    float* ws   = (float*)d_ws;
    float* agg1 = ws;                                   // 100000*128
    float* deg1 = agg1 + (size_t)N_DST1 * IN_F;         // 100000
    float* agg2 = deg1 + N_DST1;                        // 20000*256
    float* deg2 = agg2 + (size_t)N_DST2 * H_F;          // 20000
    float* h    = deg2 + N_DST2;                        // 100000*256
    const long zero_floats = (long)N_DST1 * IN_F + N_DST1 + (long)N_DST2 * H_F + N_DST2;

    // 1) zero accumulators (contiguous region)
    zero_kernel<<<4096, 256, 0, stream>>>((float4*)ws, zero_floats / 4);

    // 2) layer-1 scatter mean numerator + degree
    scatter1_kernel<<<(E1 + 7) / 8, 256, 0, stream>>>(x, src1, dst1, agg1, deg1, E1);

    // 3) layer-1 fused GEMM + bias + relu -> h
    gemm1_kernel<<<NB1 * ((MT1 + 7) / 8), 256, 0, stream>>>(
        x, agg1, deg1, Wself1, Wneigh1, b1, h);

    // 4) layer-2 scatter
    scatter2_kernel<<<(E2 + 7) / 8, 256, 0, stream>>>(h, src2, dst2, agg2, deg2, E2);

    // 5) layer-2 fused GEMM + bias -> out
    gemm2_kernel<<<((MT2 + 7) / 8), 256, 0, stream>>>(
        h, agg2, deg2, Wself2, Wneigh2, b2, out);
}